// GCN_23029614641773
// MI455X (gfx1250) — compile-verified
//
#include <hip/hip_runtime.h>

// ---------------------------------------------------------------------------
// GCN on a fixed 27-node graph, readout = node 0 only.
// Dependency-cone collapse: per sample
//   y   = x @ W0
//   agg = lrelu(y+b0) + 2*lrelu(3y+b0) + lrelu(5y+b0)   (in-degrees 1,3,3,5)
//   h2  = lrelu(agg @ W1 + b1)
//   h3  = lrelu(h2  @ W2 + b2)
//   out = h3 @ Wc + bc
// Three fp32 WMMA GEMMs (v_wmma_f32_16x16x4_f32) with fused epilogues.
// Data movement uses CDNA5 async global->LDS copies (ASYNCcnt) with
// double-buffered LDS so chunk i+1 streams in while chunk i is consumed.
// ---------------------------------------------------------------------------

typedef __attribute__((ext_vector_type(2))) float v2f;
typedef __attribute__((ext_vector_type(8))) float v8f;

#define MBLK 128
#define NBLK 64
#define KBLK 16
#define LDA  20   // padded LDS stride (floats) for A tile: 16B-aligned, bank-conflict free
#define LDB  68   // padded LDS stride (floats) for B tile

__device__ __forceinline__ float lrelu(float v) { return v > 0.0f ? v : 0.2f * v; }

// One lane-wide async copy: 16 bytes global -> LDS, tracked by ASYNCcnt.
__device__ __forceinline__ void async_b128(unsigned lds_addr, const float* gbase,
                                           unsigned byte_off) {
    asm volatile("global_load_async_to_lds_b128 %0, %1, %2"
                 :: "v"(lds_addr), "v"(byte_off), "s"(gbase)
                 : "memory");
}

// C[M,N] = epilogue(A[M,K] @ B[K,N] + bias)
// EPI==0: layer-0 fused degree-combination epilogue; EPI==1: plain leaky-relu.
template <int EPI>
__global__ __launch_bounds__(256) void gemm_wmma_f32(
    const float* __restrict__ A, const float* __restrict__ B,
    const float* __restrict__ bias, float* __restrict__ C,
    int M, int N, int K)
{
    __shared__ float As[2][MBLK * LDA];
    __shared__ float Bs[2][KBLK * LDB];

    const int tid  = threadIdx.x;
    const int wave = tid >> 5;        // 0..7
    const int lane = tid & 31;
    const int wm   = wave >> 1;       // 0..3 : 32-row slab within 128
    const int wn   = wave & 1;        // 0..1 : 32-col slab within 64
    const int half = lane >> 4;       // upper/lower 16 lanes
    const int r    = lane & 15;

    const int blockM = blockIdx.y * MBLK;
    const int blockN = blockIdx.x * NBLK;

    // LDS byte addresses of the staging buffers (flat shared addr truncates
    // to the LDS offset per the CDNA5 aperture rules).
    const unsigned asBase = (unsigned)(unsigned long long)(&As[0][0]);
    const unsigned bsBase = (unsigned)(unsigned long long)(&Bs[0][0]);

    // Per-wave async slice of one K-chunk (3 x b128 per wave):
    //   A tile 128x16 floats -> 512 lanes of 16B (2 instr/wave)
    //   B tile  16x64 floats -> 256 lanes of 16B (1 instr/wave)
    const int aSlot0 = (wave * 2 + 0) * 32 + lane;   // 0..511
    const int aSlot1 = (wave * 2 + 1) * 32 + lane;
    const int aRow0 = aSlot0 >> 2, aQ0 = aSlot0 & 3;
    const int aRow1 = aSlot1 >> 2, aQ1 = aSlot1 & 3;
    const int bSlot = wave * 32 + lane;              // 0..255
    const int bRow = bSlot >> 4, bQ = bSlot & 15;

    auto issue_chunk = [&](int kc, int buf) {
        async_b128(asBase + (unsigned)((buf * MBLK * LDA + aRow0 * LDA + aQ0 * 4) * 4),
                   A, (unsigned)(((blockM + aRow0) * K + kc + aQ0 * 4) * 4));
        async_b128(asBase + (unsigned)((buf * MBLK * LDA + aRow1 * LDA + aQ1 * 4) * 4),
                   A, (unsigned)(((blockM + aRow1) * K + kc + aQ1 * 4) * 4));
        async_b128(bsBase + (unsigned)((buf * KBLK * LDB + bRow * LDB + bQ * 4) * 4),
                   B, (unsigned)(((kc + bRow) * N + blockN + bQ * 4) * 4));
    };

    v8f acc[2][2] = {};

    const int nchunks = K / KBLK;
    issue_chunk(0, 0);

    for (int i = 0; i < nchunks; ++i) {
        const int cur = i & 1;
        const bool more = (i + 1 < nchunks);
        if (more) {
            issue_chunk((i + 1) * KBLK, cur ^ 1);
            // chunk i done when <=3 outstanding (chunk i+1's 3 still in flight)
            asm volatile("s_wait_asynccnt 0x3" ::: "memory");
        } else {
            asm volatile("s_wait_asynccnt 0x0" ::: "memory");
        }
        __syncthreads();

        const float* as = &As[cur][0];
        const float* bs = &Bs[cur][0];
#pragma unroll
        for (int kk = 0; kk < KBLK; kk += 4) {
            const int kbase = kk + 2 * half;
            // A frag (16x4 f32): lanes 0-15 hold K=0,1; lanes 16-31 hold K=2,3 (ISA 7.12.2)
            const int arow0 = wm * 32 + r;
            const int arow1 = arow0 + 16;
            v2f a0, a1, b0v, b1v;
            a0.x = as[arow0 * LDA + kbase];     a0.y = as[arow0 * LDA + kbase + 1];
            a1.x = as[arow1 * LDA + kbase];     a1.y = as[arow1 * LDA + kbase + 1];
            // B frag (4x16 f32): rows striped across lanes; upper lane-half holds rows 2,3
            const int bcol0 = wn * 32 + r;
            const int bcol1 = bcol0 + 16;
            b0v.x = bs[kbase * LDB + bcol0];    b0v.y = bs[(kbase + 1) * LDB + bcol0];
            b1v.x = bs[kbase * LDB + bcol1];    b1v.y = bs[(kbase + 1) * LDB + bcol1];

            acc[0][0] = __builtin_amdgcn_wmma_f32_16x16x4_f32(false, a0, false, b0v, (short)0, acc[0][0], false, false);
            acc[0][1] = __builtin_amdgcn_wmma_f32_16x16x4_f32(false, a0, false, b1v, (short)0, acc[0][1], false, false);
            acc[1][0] = __builtin_amdgcn_wmma_f32_16x16x4_f32(false, a1, false, b0v, (short)0, acc[1][0], false, false);
            acc[1][1] = __builtin_amdgcn_wmma_f32_16x16x4_f32(false, a1, false, b1v, (short)0, acc[1][1], false, false);
        }
        __syncthreads();
    }

    // Epilogue + store. C layout: VGPR i -> row (i + 8*half), col r within 16x16 tile.
#pragma unroll
    for (int ti = 0; ti < 2; ++ti) {
#pragma unroll
        for (int tj = 0; tj < 2; ++tj) {
            const int col = blockN + wn * 32 + tj * 16 + r;
            const float bv = bias[col];
#pragma unroll
            for (int i = 0; i < 8; ++i) {
                const int row = blockM + wm * 32 + ti * 16 + half * 8 + i;
                const float y = acc[ti][tj][i];
                float o;
                if (EPI == 0) {
                    // agg1[node1] = lrelu(1*y+b) + 2*lrelu(3*y+b) + lrelu(5*y+b)
                    o = lrelu(y + bv) + 2.0f * lrelu(3.0f * y + bv) + lrelu(5.0f * y + bv);
                } else {
                    o = lrelu(y + bv);
                }
                C[(size_t)row * N + col] = o;
            }
        }
    }
}

// out[row] = dot(H[row, 0:256], Wc) + bc   — one wave32 per row, 8 rows per block
__global__ __launch_bounds__(256) void head_kernel(
    const float* __restrict__ H, const float* __restrict__ Wc,
    const float* __restrict__ bc, float* __restrict__ out)
{
    const int row  = blockIdx.x * 8 + (threadIdx.x >> 5);
    const int lane = threadIdx.x & 31;
    const float* h = H + (size_t)row * 256;
    float s = 0.0f;
#pragma unroll
    for (int j = lane; j < 256; j += 32) s += h[j] * Wc[j];
#pragma unroll
    for (int off = 16; off > 0; off >>= 1) s += __shfl_xor(s, off, 32);
    if (lane == 0) out[row] = s + bc[0];
}

extern "C" void kernel_launch(void* const* d_in, const int* in_sizes, int n_in,
                              void* d_out, int out_size, void* d_ws, size_t ws_size,
                              hipStream_t stream)
{
    const float* x  = (const float*)d_in[0];
    const float* W0 = (const float*)d_in[1];
    const float* b0 = (const float*)d_in[2];
    const float* W1 = (const float*)d_in[3];
    const float* b1 = (const float*)d_in[4];
    const float* W2 = (const float*)d_in[5];
    const float* b2 = (const float*)d_in[6];
    const float* Wc = (const float*)d_in[7];
    const float* bc = (const float*)d_in[8];
    float* out = (float*)d_out;

    const int B = 2048, D = 1024, C0 = 1024, C1 = 512, C2 = 256;

    // Workspace: agg (8 MiB) | h2 (4 MiB) | h3 (2 MiB)  -> 14 MiB total
    float* agg = (float*)d_ws;
    float* h2  = agg + (size_t)B * C0;
    float* h3  = h2  + (size_t)B * C1;

    dim3 blk(256);
    gemm_wmma_f32<0><<<dim3(C0 / NBLK, B / MBLK), blk, 0, stream>>>(x,   W0, b0, agg, B, C0, D);
    gemm_wmma_f32<1><<<dim3(C1 / NBLK, B / MBLK), blk, 0, stream>>>(agg, W1, b1, h2,  B, C1, C0);
    gemm_wmma_f32<1><<<dim3(C2 / NBLK, B / MBLK), blk, 0, stream>>>(h2,  W2, b2, h3,  B, C2, C1);
    head_kernel<<<B / 8, 256, 0, stream>>>(h3, Wc, bc, out);
}